// Attenuation_3410204033249
// MI455X (gfx1250) — compile-verified
//
#include <hip/hip_runtime.h>

// ---------------------------------------------------------------------------
// Problem constants (from reference)
// ---------------------------------------------------------------------------
#define N_ROWS 8192
#define T_CTX  1024
#define RSQRT8 0.35355339059327373f   // 1/sqrt(KD), KD=8

typedef float v2f __attribute__((ext_vector_type(2)));
typedef float v8f __attribute__((ext_vector_type(8)));

// CDNA5 f32 WMMA: D[16x16] = A[16x4] * B[4x16] + C   (wave32, one matrix/wave)
// A frag: lanes 0-15 hold rows 0-15 with K=(0,1); lanes 16-31 K=(2,3).
// B frag: lanes 0-15 hold cols 0-15 with K=(0,1); lanes 16-31 K=(2,3).
// C/D   : VGPR r -> row r (lanes 0-15) / row r+8 (lanes 16-31), col = lane&15.
__device__ __forceinline__ v8f wmma4(v2f a, v2f b, v8f c) {
    return __builtin_amdgcn_wmma_f32_16x16x4_f32(false, a, false, b,
                                                 (short)0, c, false, false);
}

// ---------------------------------------------------------------------------
// prep_v: v[t, h*8+d] = att_coeff[t,:] @ wv + bv        ([1024,16], 64 KB)
// ---------------------------------------------------------------------------
__global__ __launch_bounds__(256) void prep_v(
    const float* __restrict__ att_coeff, const float* __restrict__ wv,
    const float* __restrict__ bv, float* __restrict__ vbuf)
{
    const int t = blockIdx.x * 256 + threadIdx.x;
    if (t >= T_CTX) return;
    float a[8];
#pragma unroll
    for (int k = 0; k < 8; ++k) a[k] = att_coeff[t * 8 + k];
#pragma unroll
    for (int j = 0; j < 16; ++j) {
        float s = bv[j];
#pragma unroll
        for (int k = 0; k < 8; ++k) s = fmaf(a[k], wv[k * 16 + j], s);
        vbuf[t * 16 + j] = s;
    }
}

// ---------------------------------------------------------------------------
// prep_q: fold wq/bq/wk/bk into per-row vectors:
//   q[h,d]   = coord1[n,:]@wq[:,h*8+d] + bq
//   qk2[h,c] = sum_d q[h,d]*wk[c,h*8+d]      (4 floats/head)
//   qbk[h]   = sum_d q[h,d]*bk[h*8+d]        (constant logit offset)
// stored as qkbuf[n][12] = {qk2[0][0..3], qk2[1][0..3], qbk[0], qbk[1], 0, 0}
// ---------------------------------------------------------------------------
__global__ __launch_bounds__(256) void prep_q(
    const float* __restrict__ coord1, const float* __restrict__ wq,
    const float* __restrict__ bq, const float* __restrict__ wk,
    const float* __restrict__ bk, float* __restrict__ qkbuf)
{
    const int n = blockIdx.x * 256 + threadIdx.x;
    if (n >= N_ROWS) return;
    float cc[4];
#pragma unroll
    for (int i = 0; i < 4; ++i) cc[i] = coord1[n * 4 + i];
    float outv[12];
#pragma unroll
    for (int h = 0; h < 2; ++h) {
        float q[8];
#pragma unroll
        for (int d = 0; d < 8; ++d) {
            const int j = h * 8 + d;
            float s = bq[j];
#pragma unroll
            for (int e = 0; e < 4; ++e) s = fmaf(cc[e], wq[e * 16 + j], s);
            q[d] = s;
        }
#pragma unroll
        for (int e = 0; e < 4; ++e) {
            float s = 0.f;
#pragma unroll
            for (int d = 0; d < 8; ++d) s = fmaf(q[d], wk[e * 16 + h * 8 + d], s);
            outv[h * 4 + e] = s;
        }
        float sb = 0.f;
#pragma unroll
        for (int d = 0; d < 8; ++d) sb = fmaf(q[d], bk[h * 8 + d], sb);
        outv[8 + h] = sb;
    }
    outv[10] = 0.f; outv[11] = 0.f;
#pragma unroll
    for (int i = 0; i < 12; ++i) qkbuf[n * 12 + i] = outv[i];
}

// ---------------------------------------------------------------------------
// attn_kernel: one wave per row n, lanes stride over T (t = lane + 32*i).
// Streams local_coords (128 MB) exactly once; flash-style online softmax per
// head; v table lives in LDS padded to stride 20 dwords (<=2-way conflicts).
// Result attn[n, h*8+d] -> attnbuf [N,16].
// ---------------------------------------------------------------------------
__device__ __forceinline__ void combine_head(float& m, float& l,
                                             float4& a0, float4& a1, int off) {
    const float mo = __shfl_xor(m, off, 32);
    const float lo = __shfl_xor(l, off, 32);
    float4 b0, b1;
    b0.x = __shfl_xor(a0.x, off, 32); b0.y = __shfl_xor(a0.y, off, 32);
    b0.z = __shfl_xor(a0.z, off, 32); b0.w = __shfl_xor(a0.w, off, 32);
    b1.x = __shfl_xor(a1.x, off, 32); b1.y = __shfl_xor(a1.y, off, 32);
    b1.z = __shfl_xor(a1.z, off, 32); b1.w = __shfl_xor(a1.w, off, 32);
    const float nm = fmaxf(m, mo);
    const float ss = __expf(m - nm);
    const float so = __expf(mo - nm);
    l = l * ss + lo * so;
    a0.x = a0.x * ss + b0.x * so; a0.y = a0.y * ss + b0.y * so;
    a0.z = a0.z * ss + b0.z * so; a0.w = a0.w * ss + b0.w * so;
    a1.x = a1.x * ss + b1.x * so; a1.y = a1.y * ss + b1.y * so;
    a1.z = a1.z * ss + b1.z * so; a1.w = a1.w * ss + b1.w * so;
    m = nm;
}

__global__ __launch_bounds__(256) void attn_kernel(
    const float4* __restrict__ lc,          // [N][T] of float4 (CD=4)
    const unsigned char* __restrict__ mask, // [N][T] bool bytes
    const float* __restrict__ qkbuf,        // [N][12]
    const float* __restrict__ vbuf,         // [T][16]
    float* __restrict__ attnbuf)            // [N][16]
{
    __shared__ float vlds[T_CTX * 20];      // 80 KB, stride-20 pad for banks
    const int tid = threadIdx.x;
    for (int t = tid; t < T_CTX; t += 256) {
        const float4* vp = reinterpret_cast<const float4*>(vbuf + t * 16);
        float4* dp = reinterpret_cast<float4*>(&vlds[t * 20]);
        dp[0] = vp[0]; dp[1] = vp[1]; dp[2] = vp[2]; dp[3] = vp[3];
    }
    __syncthreads();

    const int wave = tid >> 5, lane = tid & 31;
    const int n = blockIdx.x * 8 + wave;

    const float* qp = qkbuf + n * 12;
    const float qk0x = qp[0], qk0y = qp[1], qk0z = qp[2], qk0w = qp[3];
    const float qk1x = qp[4], qk1y = qp[5], qk1z = qp[6], qk1w = qp[7];
    const float qb0 = qp[8], qb1 = qp[9];

    const float4* lcrow = lc + (size_t)n * T_CTX;
    const unsigned char* mrow = mask + (size_t)n * T_CTX;

    float m0 = -1e30f, m1 = -1e30f, l0 = 0.f, l1 = 0.f;
    float4 A00 = {0, 0, 0, 0}, A01 = {0, 0, 0, 0};
    float4 A10 = {0, 0, 0, 0}, A11 = {0, 0, 0, 0};

#pragma unroll 4
    for (int i = 0; i < 32; ++i) {
        const int t = lane + (i << 5);
        const float4 c = lcrow[t];               // 512B/wave, fully coalesced
        const bool ok = mrow[t] != 0;
        if (i < 24) __builtin_prefetch(lcrow + t + 256, 0, 0); // global_prefetch
        if (ok) {
            const float lg0 = fmaf(qk0x, c.x, fmaf(qk0y, c.y,
                              fmaf(qk0z, c.z, fmaf(qk0w, c.w, qb0)))) * RSQRT8;
            const float lg1 = fmaf(qk1x, c.x, fmaf(qk1y, c.y,
                              fmaf(qk1z, c.z, fmaf(qk1w, c.w, qb1)))) * RSQRT8;
            const float* vt = &vlds[t * 20];
            const float4 v0 = *(const float4*)(vt);
            const float4 v1 = *(const float4*)(vt + 4);
            const float4 v2 = *(const float4*)(vt + 8);
            const float4 v3 = *(const float4*)(vt + 12);
            // head 0 (rare-rescale online softmax)
            if (lg0 > m0) {
                const float s = __expf(m0 - lg0);
                l0 = fmaf(l0, s, 1.f);
                A00.x = fmaf(A00.x, s, v0.x); A00.y = fmaf(A00.y, s, v0.y);
                A00.z = fmaf(A00.z, s, v0.z); A00.w = fmaf(A00.w, s, v0.w);
                A01.x = fmaf(A01.x, s, v1.x); A01.y = fmaf(A01.y, s, v1.y);
                A01.z = fmaf(A01.z, s, v1.z); A01.w = fmaf(A01.w, s, v1.w);
                m0 = lg0;
            } else {
                const float p = __expf(lg0 - m0);
                l0 += p;
                A00.x = fmaf(p, v0.x, A00.x); A00.y = fmaf(p, v0.y, A00.y);
                A00.z = fmaf(p, v0.z, A00.z); A00.w = fmaf(p, v0.w, A00.w);
                A01.x = fmaf(p, v1.x, A01.x); A01.y = fmaf(p, v1.y, A01.y);
                A01.z = fmaf(p, v1.z, A01.z); A01.w = fmaf(p, v1.w, A01.w);
            }
            // head 1
            if (lg1 > m1) {
                const float s = __expf(m1 - lg1);
                l1 = fmaf(l1, s, 1.f);
                A10.x = fmaf(A10.x, s, v2.x); A10.y = fmaf(A10.y, s, v2.y);
                A10.z = fmaf(A10.z, s, v2.z); A10.w = fmaf(A10.w, s, v2.w);
                A11.x = fmaf(A11.x, s, v3.x); A11.y = fmaf(A11.y, s, v3.y);
                A11.z = fmaf(A11.z, s, v3.z); A11.w = fmaf(A11.w, s, v3.w);
                m1 = lg1;
            } else {
                const float p = __expf(lg1 - m1);
                l1 += p;
                A10.x = fmaf(p, v2.x, A10.x); A10.y = fmaf(p, v2.y, A10.y);
                A10.z = fmaf(p, v2.z, A10.z); A10.w = fmaf(p, v2.w, A10.w);
                A11.x = fmaf(p, v3.x, A11.x); A11.y = fmaf(p, v3.y, A11.y);
                A11.z = fmaf(p, v3.z, A11.z); A11.w = fmaf(p, v3.w, A11.w);
            }
        }
    }

    // cross-lane butterfly merge of (m, l, acc)
#pragma unroll
    for (int off = 16; off > 0; off >>= 1) {
        combine_head(m0, l0, A00, A01, off);
        combine_head(m1, l1, A10, A11, off);
    }

    if (lane == 0) {
        const float i0 = (l0 > 0.f) ? 1.f / l0 : 0.f;
        const float i1 = (l1 > 0.f) ? 1.f / l1 : 0.f;
        float4* op = reinterpret_cast<float4*>(attnbuf + (size_t)n * 16);
        float4 o0 = {A00.x * i0, A00.y * i0, A00.z * i0, A00.w * i0};
        float4 o1 = {A01.x * i0, A01.y * i0, A01.z * i0, A01.w * i0};
        float4 o2 = {A10.x * i1, A10.y * i1, A10.z * i1, A10.w * i1};
        float4 o3 = {A11.x * i1, A11.y * i1, A11.z * i1, A11.w * i1};
        op[0] = o0; op[1] = o1; op[2] = o2; op[3] = o3;
    }
}

// ---------------------------------------------------------------------------
// mlp_kernel: one wave owns a 16-row tile. All GEMMs via V_WMMA_F32_16X16X4.
//   att = exp(-(attn@wo + bo))              K=16 -> 4 WMMAs
//   h1  = tanh([c1,c2,att]@w1 + b1)         K=24, Nout=64 -> 24 WMMAs
//   h2  = tanh(h1@w2 + b2)                  K=64, Nout=64 -> 64 WMMAs
//   out = tanh(h2@wout + bout)              K=64, Nout=32 -> 32 WMMAs
// Activations round-trip through bank-padded LDS (strides 34 / 66).
// ---------------------------------------------------------------------------
__global__ __launch_bounds__(128) void mlp_kernel(
    const float* __restrict__ attnbuf, const float* __restrict__ coord1,
    const float* __restrict__ coord2,
    const float* __restrict__ wo, const float* __restrict__ bo,
    const float* __restrict__ w1, const float* __restrict__ b1,
    const float* __restrict__ w2, const float* __restrict__ b2,
    const float* __restrict__ wout, const float* __restrict__ bout,
    float* __restrict__ out)
{
    constexpr int ACT_STRIDE = 34;                 // 24 used cols, padded
    constexpr int H_STRIDE   = 66;                 // 64 used cols, padded
    constexpr int WAVE_LDS   = 16 * ACT_STRIDE + 2 * 16 * H_STRIDE; // 2656 f
    __shared__ float lds[4 * WAVE_LDS];            // 4 waves/block, ~42.5 KB

    const int lane = threadIdx.x & 31;
    const int wave = threadIdx.x >> 5;
    float* act = lds + wave * WAVE_LDS;            // [16][34]
    float* h1  = act + 16 * ACT_STRIDE;            // [16][66]
    float* h2  = h1 + 16 * H_STRIDE;               // [16][66]

    const int r0   = (blockIdx.x * 4 + wave) * 16;
    const int half = lane >> 4;                    // K sub-pair select
    const int lx   = lane & 15;                    // row (A) / col (B,C)

    // ---- att = exp(-(attn @ wo + bo)) ----
    v8f c0 = {};
#pragma unroll
    for (int kc = 0; kc < 4; ++kc) {
        const int ka = kc * 4 + half * 2;
        const float* ap = attnbuf + (size_t)(r0 + lx) * 16 + ka;
        v2f a; a.x = ap[0]; a.y = ap[1];
        const float* bp = wo + ka * 16 + lx;
        v2f b; b.x = bp[0]; b.y = bp[16];
        c0 = wmma4(a, b, c0);
    }
    {
        const float bv = bo[lx];
#pragma unroll
        for (int r = 0; r < 8; ++r)
            act[(r + half * 8) * ACT_STRIDE + 8 + lx] = __expf(-(c0[r] + bv));
    }
    // coords -> act cols 0..7
    for (int i = lane; i < 64; i += 32) {
        const int rr = i >> 2, cc = i & 3;
        act[rr * ACT_STRIDE + cc]     = coord1[(size_t)(r0 + rr) * 4 + cc];
        act[rr * ACT_STRIDE + 4 + cc] = coord2[(size_t)(r0 + rr) * 4 + cc];
    }
    __syncthreads();

    // ---- layer 1: [16,24] @ [24,64] ----
#pragma unroll
    for (int nt = 0; nt < 4; ++nt) {
        v8f acc = {};
#pragma unroll
        for (int kc = 0; kc < 6; ++kc) {
            const int ka = kc * 4 + half * 2;
            const v2f a = *(const v2f*)&act[lx * ACT_STRIDE + ka];
            const float* bp = w1 + ka * 64 + nt * 16 + lx;
            v2f b; b.x = bp[0]; b.y = bp[64];
            acc = wmma4(a, b, acc);
        }
        const float bv = b1[nt * 16 + lx];
#pragma unroll
        for (int r = 0; r < 8; ++r)
            h1[(r + half * 8) * H_STRIDE + nt * 16 + lx] = tanhf(acc[r] + bv);
    }
    __syncthreads();

    // ---- layer 2: [16,64] @ [64,64] ----
#pragma unroll
    for (int nt = 0; nt < 4; ++nt) {
        v8f acc = {};
#pragma unroll
        for (int kc = 0; kc < 16; ++kc) {
            const int ka = kc * 4 + half * 2;
            const v2f a = *(const v2f*)&h1[lx * H_STRIDE + ka];
            const float* bp = w2 + ka * 64 + nt * 16 + lx;
            v2f b; b.x = bp[0]; b.y = bp[64];
            acc = wmma4(a, b, acc);
        }
        const float bv = b2[nt * 16 + lx];
#pragma unroll
        for (int r = 0; r < 8; ++r)
            h2[(r + half * 8) * H_STRIDE + nt * 16 + lx] = tanhf(acc[r] + bv);
    }
    __syncthreads();

    // ---- layer 3: [16,64] @ [64,32] -> out ----
#pragma unroll
    for (int nt = 0; nt < 2; ++nt) {
        v8f acc = {};
#pragma unroll
        for (int kc = 0; kc < 16; ++kc) {
            const int ka = kc * 4 + half * 2;
            const v2f a = *(const v2f*)&h2[lx * H_STRIDE + ka];
            const float* bp = wout + ka * 32 + nt * 16 + lx;
            v2f b; b.x = bp[0]; b.y = bp[32];
            acc = wmma4(a, b, acc);
        }
        const float bv = bout[nt * 16 + lx];
#pragma unroll
        for (int r = 0; r < 8; ++r)
            out[(size_t)(r0 + r + half * 8) * 32 + nt * 16 + lx] =
                tanhf(acc[r] + bv);
    }
}

// ---------------------------------------------------------------------------
// kernel_launch
// inputs (setup_inputs order): coord1, coord2, att_coeff, local_coords, mask,
// wq, bq, wk, bk, wv, bv, wo, bo, w1, b1, w2, b2, wout, bout
// ---------------------------------------------------------------------------
extern "C" void kernel_launch(void* const* d_in, const int* in_sizes, int n_in,
                              void* d_out, int out_size, void* d_ws,
                              size_t ws_size, hipStream_t stream) {
    const float* coord1       = (const float*)d_in[0];
    const float* coord2       = (const float*)d_in[1];
    const float* att_coeff    = (const float*)d_in[2];
    const float* local_coords = (const float*)d_in[3];
    const unsigned char* mask = (const unsigned char*)d_in[4];
    const float* wq   = (const float*)d_in[5];
    const float* bq   = (const float*)d_in[6];
    const float* wk   = (const float*)d_in[7];
    const float* bk   = (const float*)d_in[8];
    const float* wv   = (const float*)d_in[9];
    const float* bv   = (const float*)d_in[10];
    const float* wo   = (const float*)d_in[11];
    const float* bo   = (const float*)d_in[12];
    const float* w1   = (const float*)d_in[13];
    const float* b1   = (const float*)d_in[14];
    const float* w2   = (const float*)d_in[15];
    const float* b2   = (const float*)d_in[16];
    const float* wout = (const float*)d_in[17];
    const float* bout = (const float*)d_in[18];
    float* out = (float*)d_out;

    // workspace layout (floats): vbuf[T*16] | qkbuf[N*12] | attnbuf[N*16]
    float* ws      = (float*)d_ws;
    float* vbuf    = ws;
    float* qkbuf   = vbuf + T_CTX * 16;
    float* attnbuf = qkbuf + N_ROWS * 12;
    (void)in_sizes; (void)n_in; (void)out_size; (void)ws_size;

    prep_v<<<(T_CTX + 255) / 256, 256, 0, stream>>>(att_coeff, wv, bv, vbuf);
    prep_q<<<(N_ROWS + 255) / 256, 256, 0, stream>>>(coord1, wq, bq, wk, bk,
                                                     qkbuf);
    attn_kernel<<<N_ROWS / 8, 256, 0, stream>>>(
        (const float4*)local_coords, mask, qkbuf, vbuf, attnbuf);
    mlp_kernel<<<N_ROWS / 64, 128, 0, stream>>>(
        attnbuf, coord1, coord2, wo, bo, w1, b1, w2, b2, wout, bout, out);
}